// CombinedHiddenDecoder_26800595927063
// MI455X (gfx1250) — compile-verified
//
#include <hip/hip_runtime.h>
#include <hip/hip_bf16.h>

typedef __attribute__((ext_vector_type(2))) float v2f;
typedef __attribute__((ext_vector_type(8))) float v8f;

// LDS padding to kill bank conflicts (64 banks x 4B):
//  A-tile row stride = K+4  (== 4 mod 64 -> lanes m=0..15 hit banks 4m+*, disjoint)
//  W     row stride = 72    (== 8 mod 64 -> half-wave kb vs kb+2 bank ranges disjoint)
#define WPAD 72

// ---------------------------------------------------------------------------
// Utility kernels
// ---------------------------------------------------------------------------
__global__ void fill_f32(float* __restrict__ p, float v, long n) {
  long i = (long)blockIdx.x * blockDim.x + threadIdx.x;
  if (i < n) p[i] = v;
}

// deg[col[e]] += 1 for every edge (deg pre-initialized to 1.0 = self loop)
__global__ void count_deg(const int* __restrict__ col, float* __restrict__ deg, int nE) {
  int e = blockIdx.x * blockDim.x + threadIdx.x;
  if (e < nE) unsafeAtomicAdd(&deg[col[e]], 1.0f);
}

// in-place deg -> rsqrt(deg)
__global__ void make_dinv(float* __restrict__ d, int n) {
  int i = blockIdx.x * blockDim.x + threadIdx.x;
  if (i < n) {
    float v = d[i];
    d[i] = (v > 0.0f) ? rsqrtf(v) : 0.0f;
  }
}

// ---------------------------------------------------------------------------
// WMMA GEMM:  Y[N x 64] = (X[N x K] @ W[K x 64]) * dinv[row]   (row-scaled)
// Block = 128 threads = 4 waves; block computes a 16 x 64 tile.
// Wave w computes columns [16w, 16w+16) with V_WMMA_F32_16X16X4_F32.
// ---------------------------------------------------------------------------
template <int K>
__global__ void __launch_bounds__(128)
gcn_gemm_wmma(const float* __restrict__ X, const float* __restrict__ W,
              const float* __restrict__ dinv, float* __restrict__ Y, int nRows) {
  __shared__ float wlds[K * WPAD];      // W staged once per block (padded rows)
  __shared__ float alds[16 * (K + 4)];  // 16-row A tile (padded rows)

  const int tid  = threadIdx.x;
  const int wave = tid >> 5;
  const int lane = tid & 31;
  const int tile = blockIdx.x * 16;

  // Stage W (K x 64, rows contiguous & 16B aligned) as float4 -> padded LDS
  for (int i = tid * 4; i < K * 64; i += 128 * 4) {
    const int k = i >> 6;        // row
    const int c = i & 63;        // col (multiple of 4)
    *reinterpret_cast<float4*>(&wlds[k * WPAD + c]) =
        *reinterpret_cast<const float4*>(W + i);
  }
  // Stage 16-row A tile as float4 -> padded LDS
  for (int i = tid * 4; i < 16 * K; i += 128 * 4) {
    const int r = i / K;
    const int c = i % K;         // multiple of 4 (K is a multiple of 4)
    int gr = tile + r;
    if (gr >= nRows) gr = nRows - 1;  // clamp (N is a multiple of 16 anyway)
    *reinterpret_cast<float4*>(&alds[r * (K + 4) + c]) =
        *reinterpret_cast<const float4*>(X + (long)gr * K + c);
  }
  __syncthreads();

  const int m  = lane & 15;                // A-matrix row within tile
  const int kq = (lane >> 4) * 2;          // K sub-offset per ISA 16x4 fp32 layout
  const int n  = (lane & 15) + wave * 16;  // B/C/D column

  v8f c = {};
#pragma unroll
  for (int kk = 0; kk < K; kk += 4) {
    const int kb = kk + kq;
    v2f a, b;
    a.x = alds[m * (K + 4) + kb];
    a.y = alds[m * (K + 4) + kb + 1];
    b.x = wlds[kb * WPAD + n];
    b.y = wlds[(kb + 1) * WPAD + n];
    // D = A(16x4) * B(4x16) + C ; fp32 in, fp32 accumulate
    c = __builtin_amdgcn_wmma_f32_16x16x4_f32(false, a, false, b, (short)0, c,
                                              false, false);
  }

  // C/D layout: VGPR v, lane l -> M = v + 8*(l>>4), N = l&15 (+ wave offset)
  const int mrow0 = (lane >> 4) * 8;
#pragma unroll
  for (int v = 0; v < 8; ++v) {
    const int gr = tile + mrow0 + v;
    if (gr < nRows) Y[(long)gr * 64 + n] = c[v] * dinv[gr];
  }
}

// ---------------------------------------------------------------------------
// Edge scatter: acc[col[e], f] += y[row[e], f]   (y already scaled by dinv[row])
// One thread handles one edge x 4 features (float4 gather, 4 hw f32 atomics).
// Node features (25.6 MB) are L2-resident on MI455X (192 MB L2), so both the
// gather and the atomics run at L2 bandwidth, not HBM.
// ---------------------------------------------------------------------------
__global__ void __launch_bounds__(256)
scatter_edges(const int* __restrict__ row, const int* __restrict__ col,
              const float* __restrict__ y, float* __restrict__ acc, long nWork) {
  long t = (long)blockIdx.x * 256 + threadIdx.x;
  if (t >= nWork) return;
  const int e = (int)(t >> 4);
  const int f = (int)(t & 15) << 2;
  const int r = row[e];
  const int cd = col[e];
  const float4 v = *reinterpret_cast<const float4*>(y + (long)r * 64 + f);
  float* dst = acc + (long)cd * 64 + f;
  unsafeAtomicAdd(dst + 0, v.x);
  unsafeAtomicAdd(dst + 1, v.y);
  unsafeAtomicAdd(dst + 2, v.z);
  unsafeAtomicAdd(dst + 3, v.w);
}

// ---------------------------------------------------------------------------
// Finalize: out[i, f] = dinv[i] * (acc[i,f] + y[i,f]) + b[f]
//  (y term = self loop: y[i]*dinv[i] = xW[i]*dinv[i]^2).  ostride lets us
//  write directly into the concat buffer (stride 128, col offset via pointer).
// ---------------------------------------------------------------------------
__global__ void __launch_bounds__(256)
finalize_layer(const float* __restrict__ acc, const float* __restrict__ y,
               const float* __restrict__ dinv, const float* __restrict__ bias,
               float* __restrict__ out, int ostride, int nNodes) {
  long t = (long)blockIdx.x * 256 + threadIdx.x;  // nNodes*16 work items
  if (t >= (long)nNodes * 16) return;
  const int i = (int)(t >> 4);
  const int f = (int)(t & 15) << 2;
  const float d = dinv[i];
  const float4 a = *reinterpret_cast<const float4*>(acc + (long)i * 64 + f);
  const float4 s = *reinterpret_cast<const float4*>(y + (long)i * 64 + f);
  const float4 b = *reinterpret_cast<const float4*>(bias + f);
  float4 o;
  o.x = d * (a.x + s.x) + b.x;
  o.y = d * (a.y + s.y) + b.y;
  o.z = d * (a.z + s.z) + b.z;
  o.w = d * (a.w + s.w) + b.w;
  *reinterpret_cast<float4*>(out + (long)i * ostride + f) = o;
}

// ---------------------------------------------------------------------------
// Orchestration
// ---------------------------------------------------------------------------
extern "C" void kernel_launch(void* const* d_in, const int* in_sizes, int n_in,
                              void* d_out, int out_size, void* d_ws, size_t ws_size,
                              hipStream_t stream) {
  const float* latent    = (const float*)d_in[0];
  const float* condition = (const float*)d_in[1];
  const int*   ei        = (const int*)d_in[2];
  const float* W1 = (const float*)d_in[3];
  const float* b1 = (const float*)d_in[4];
  const float* W2 = (const float*)d_in[5];
  const float* b2 = (const float*)d_in[6];
  const float* W3 = (const float*)d_in[7];
  const float* b3 = (const float*)d_in[8];
  const float* W4 = (const float*)d_in[9];
  const float* b4 = (const float*)d_in[10];

  const int N = in_sizes[0] / 64;   // 100000
  const int E = in_sizes[2] / 2;    // 1600000
  const int* rowp = ei;
  const int* colp = ei + E;

  // Workspace layout (256B aligned slices)
  char* ws = (char*)d_ws;
  size_t off = 0;
  float* dinv = (float*)(ws + off);
  off += ((size_t)N * 4 + 255) & ~(size_t)255;
  float* ybuf = (float*)(ws + off);
  off += (size_t)N * 64 * 4;        // multiple of 256
  float* acc  = (float*)(ws + off);
  off += (size_t)N * 64 * 4;
  float* hbuf = (float*)(ws + off); // N x 128 concat buffer (also reused as h3)

  const long accElems = (long)N * 64;
  const long edgeWork = (long)E * 16;
  const dim3 gemmGrid((N + 15) / 16);
  const dim3 gemmBlk(128);
  const dim3 fillN((N + 255) / 256);
  const dim3 fillAcc((accElems + 255) / 256);
  const dim3 edgeGrid((edgeWork + 255) / 256);
  const dim3 finGrid(((long)N * 16 + 255) / 256);

  // Degree normalization (shared by all 4 layers)
  fill_f32<<<fillN, 256, 0, stream>>>(dinv, 1.0f, N);
  count_deg<<<(E + 255) / 256, 256, 0, stream>>>(colp, dinv, E);
  make_dinv<<<fillN, 256, 0, stream>>>(dinv, N);

  // Layer 1: latent @ W1 -> h[:, 0:64]
  gcn_gemm_wmma<64><<<gemmGrid, gemmBlk, 0, stream>>>(latent, W1, dinv, ybuf, N);
  fill_f32<<<fillAcc, 256, 0, stream>>>(acc, 0.0f, accElems);
  scatter_edges<<<edgeGrid, 256, 0, stream>>>(rowp, colp, ybuf, acc, edgeWork);
  finalize_layer<<<finGrid, 256, 0, stream>>>(acc, ybuf, dinv, b1, hbuf, 128, N);

  // Layer 2: condition @ W2 -> h[:, 64:128]
  gcn_gemm_wmma<32><<<gemmGrid, gemmBlk, 0, stream>>>(condition, W2, dinv, ybuf, N);
  fill_f32<<<fillAcc, 256, 0, stream>>>(acc, 0.0f, accElems);
  scatter_edges<<<edgeGrid, 256, 0, stream>>>(rowp, colp, ybuf, acc, edgeWork);
  finalize_layer<<<finGrid, 256, 0, stream>>>(acc, ybuf, dinv, b2, hbuf + 64, 128, N);

  // Layer 3: h[N x 128] @ W3 -> h3 (reuse hbuf front, stride 64; h is dead
  // after this GEMM reads it)
  gcn_gemm_wmma<128><<<gemmGrid, gemmBlk, 0, stream>>>(hbuf, W3, dinv, ybuf, N);
  fill_f32<<<fillAcc, 256, 0, stream>>>(acc, 0.0f, accElems);
  scatter_edges<<<edgeGrid, 256, 0, stream>>>(rowp, colp, ybuf, acc, edgeWork);
  finalize_layer<<<finGrid, 256, 0, stream>>>(acc, ybuf, dinv, b3, hbuf, 64, N);

  // Layer 4: h3 @ W4 -> out
  gcn_gemm_wmma<64><<<gemmGrid, gemmBlk, 0, stream>>>(hbuf, W4, dinv, ybuf, N);
  fill_f32<<<fillAcc, 256, 0, stream>>>(acc, 0.0f, accElems);
  scatter_edges<<<edgeGrid, 256, 0, stream>>>(rowp, colp, ybuf, acc, edgeWork);
  finalize_layer<<<finGrid, 256, 0, stream>>>(acc, ybuf, dinv, b4, (float*)d_out, 64, N);
}